// YoloLoss_51307679318799
// MI455X (gfx1250) — compile-verified
//
#include <hip/hip_runtime.h>

// ---------------------------------------------------------------------------
// YOLO loss for MI455X (gfx1250, wave32).
// Memory-bound streaming reduction:
//   - coalesced b128 staging of pred/cls tiles into LDS (async-to-LDS path
//     when the gfx1250 builtins are available)
//   - per-wave reduction of the 4 loss accumulators via V_WMMA_F32_16X16X4_F32
//   - deterministic two-stage (block partials -> finalize) reduction
// ---------------------------------------------------------------------------

#ifndef __has_builtin
#define __has_builtin(x) 0
#endif

#if __has_builtin(__builtin_amdgcn_wmma_f32_16x16x4_f32)
#define HAVE_WMMA_F32X4 1
#endif

#if __has_builtin(__builtin_amdgcn_global_load_async_to_lds_b128) && \
    __has_builtin(__builtin_amdgcn_s_wait_asynccnt)
#define HAVE_ASYNC_LDS 1
#endif

typedef __attribute__((ext_vector_type(2))) float v2f;
typedef __attribute__((ext_vector_type(8))) float v8f;

#define SGRID_INV (1.0f / 14.0f)
#define L_COORD 5.0f
#define L_NOOBJ 0.5f

constexpr int CPB    = 256;          // cells per block == blockDim.x
constexpr int PRED_F = 30;           // B*5 + C = 2*5 + 20
constexpr int CLS_F  = 20;
constexpr int PRED_V4 = CPB * PRED_F / 4;  // 1920 float4 per tile
constexpr int CLS_V4  = CPB * CLS_F / 4;   // 1280 float4 per tile

#if defined(HAVE_ASYNC_LDS)
// Builtin prototype (from hipcc diagnostic): param0 is
//   'int __vector(4) __attribute__((address_space(1)))*'  (global, AS1)
// so pass 16-byte int-vector pointers in the right address spaces.
typedef int v4i __attribute__((vector_size(16)));
typedef __attribute__((address_space(1))) v4i as1_v4i;
typedef __attribute__((address_space(3))) v4i as3_v4i;

// Generic->AS1 keeps the full 64-bit value; generic->AS3 keeps addr[31:0]
// (per ISA: the LDS flat aperture truncates to the 32-bit LDS offset).
__device__ __forceinline__ void async_b128(const void* gsrc, void* ldst) {
  __builtin_amdgcn_global_load_async_to_lds_b128(
      (as1_v4i*)(unsigned long long)gsrc,
      (as3_v4i*)(unsigned int)(unsigned long long)ldst,
      0, 0);
}
#endif

// ---------------------------------------------------------------------------
// Reduce 4 per-lane f32 accumulators across the 32 lanes of a wave.
// WMMA path: A(16x4) holds (c0,c1) per lane; B(4x16) is a one-hot selector
// with B[0][0]=B[2][0]=1 and B[1][1]=B[3][1]=1, so
//   D[m][0] = c0(m) + c0(m+16),  D[m][1] = c1(m) + c1(m+16).
// Summing the 8 C/D VGPRs per lane leaves the halves of the column sums in
// lanes {0,16} (comp0) and {1,17} (comp1); two shuffles finish it.
// ---------------------------------------------------------------------------
__device__ __forceinline__ void wave_reduce4(float c0, float c1, float c2, float c3,
                                             float& r0, float& r1, float& r2, float& r3) {
#if defined(HAVE_WMMA_F32X4)
  const int lane = threadIdx.x & 31;
  v2f a01 = {c0, c1};
  v2f a23 = {c2, c3};
  v2f bsel;
  bsel.x = (lane == 0 || lane == 16) ? 1.0f : 0.0f;
  bsel.y = (lane == 1 || lane == 17) ? 1.0f : 0.0f;
  v8f zero = {0.f, 0.f, 0.f, 0.f, 0.f, 0.f, 0.f, 0.f};
  v8f d01 = __builtin_amdgcn_wmma_f32_16x16x4_f32(false, a01, false, bsel,
                                                  (short)0, zero, false, false);
  v8f d23 = __builtin_amdgcn_wmma_f32_16x16x4_f32(false, a23, false, bsel,
                                                  (short)0, zero, false, false);
  float s01 = ((d01[0] + d01[1]) + (d01[2] + d01[3])) +
              ((d01[4] + d01[5]) + (d01[6] + d01[7]));
  float s23 = ((d23[0] + d23[1]) + (d23[2] + d23[3])) +
              ((d23[4] + d23[5]) + (d23[6] + d23[7]));
  r0 = __shfl(s01, 0, 32) + __shfl(s01, 16, 32);
  r1 = __shfl(s01, 1, 32) + __shfl(s01, 17, 32);
  r2 = __shfl(s23, 0, 32) + __shfl(s23, 16, 32);
  r3 = __shfl(s23, 1, 32) + __shfl(s23, 17, 32);
#else
  #pragma unroll
  for (int off = 16; off > 0; off >>= 1) {
    c0 += __shfl_xor(c0, off, 32);
    c1 += __shfl_xor(c1, off, 32);
    c2 += __shfl_xor(c2, off, 32);
    c3 += __shfl_xor(c3, off, 32);
  }
  r0 = c0; r1 = c1; r2 = c2; r3 = c3;
#endif
}

// ---------------------------------------------------------------------------
// Main kernel: one cell per thread, 256 cells per block, grid sized exactly
// (no partial blocks -> EXEC all ones at WMMA). Emits per-block partials:
//   partials[blk] = {reg_raw, containing, noobj_raw, cls}
// ---------------------------------------------------------------------------
__global__ void __launch_bounds__(CPB)
yolo_main(const float* __restrict__ pred,
          const float* __restrict__ tbox,
          const float* __restrict__ tcls,
          const unsigned char* __restrict__ mask,   // jnp.bool_ -> 1 byte/elem
          float* __restrict__ partials) {
  __shared__ __align__(16) float sPred[CPB * PRED_F];  // 30 KB
  __shared__ __align__(16) float sCls[CPB * CLS_F];    // 20 KB
  __shared__ float sRed[8 * 4];

  const int tid  = threadIdx.x;
  const int base = blockIdx.x * CPB;

  // ---- stage pred + target_cls tiles, fully coalesced b128 ----
  const float4* gp4 = (const float4*)(pred + (size_t)base * PRED_F);
  const float4* gc4 = (const float4*)(tcls + (size_t)base * CLS_F);
#if defined(HAVE_ASYNC_LDS)
  #pragma unroll
  for (int j = 0; j < 8; ++j) {
    int idx = j * CPB + tid;
    if (idx < PRED_V4) async_b128(gp4 + idx, ((float4*)sPred) + idx);
  }
  #pragma unroll
  for (int j = 0; j < 5; ++j) {
    int idx = j * CPB + tid;
    async_b128(gc4 + idx, ((float4*)sCls) + idx);
  }
  __builtin_amdgcn_s_wait_asynccnt(0);
#else
  #pragma unroll
  for (int j = 0; j < 8; ++j) {
    int idx = j * CPB + tid;
    if (idx < PRED_V4) ((float4*)sPred)[idx] = gp4[idx];
  }
  #pragma unroll
  for (int j = 0; j < 5; ++j) {
    int idx = j * CPB + tid;
    ((float4*)sCls)[idx] = gc4[idx];
  }
#endif
  __syncthreads();

  const int cell = base + tid;
  const float* P = &sPred[tid * PRED_F];
  const float* C = &sCls[tid * CLS_F];
  const float4 tb = ((const float4*)tbox)[cell];
  const float mf  = mask[cell] ? 1.0f : 0.0f;

  // ---- class loss: mask * sum_c (tcls - pcls)^2 ----
  float cls = 0.0f;
  #pragma unroll
  for (int c = 0; c < CLS_F; ++c) {
    float d = C[c] - P[10 + c];
    cls = fmaf(d, d, cls);
  }
  cls *= mf;

  // ---- no-object confidence loss (raw; * L_NOOBJ at finalize) ----
  float conf0 = P[4], conf1 = P[9];
  float noobj = (1.0f - mf) * (conf0 * conf0 + conf1 * conf1);

  // ---- IoU of both pred boxes vs target ----
  float tx = tb.x * SGRID_INV, ty = tb.y * SGRID_INV;
  float t_x1 = tx - 0.5f * tb.z, t_y1 = ty - 0.5f * tb.w;
  float t_x2 = tx + 0.5f * tb.z, t_y2 = ty + 0.5f * tb.w;
  float area_t = tb.z * tb.w;

  float iou[2];
  #pragma unroll
  for (int b = 0; b < 2; ++b) {
    const float* q = P + b * 5;
    float px = q[0] * SGRID_INV, py = q[1] * SGRID_INV;
    float pw = q[2], ph = q[3];
    float p_x1 = px - 0.5f * pw, p_y1 = py - 0.5f * ph;
    float p_x2 = px + 0.5f * pw, p_y2 = py + 0.5f * ph;
    float wx = fmaxf(fminf(t_x2, p_x2) - fmaxf(t_x1, p_x1), 0.0f);
    float wy = fmaxf(fminf(t_y2, p_y2) - fmaxf(t_y1, p_y1), 0.0f);
    float inter = wx * wy;
    iou[b] = inter / (area_t + pw * ph - inter);
  }
  const bool second  = iou[1] > iou[0];          // argmax, ties -> box 0
  const float best_iou = second ? iou[1] : iou[0];
  const float* Bst = P + (second ? 5 : 0);

  // ---- regression loss (raw; * L_COORD at finalize) ----
  float dx = Bst[0] - tb.x, dy = Bst[1] - tb.y;
  float dw = __builtin_sqrtf(Bst[2]) - __builtin_sqrtf(tb.z);
  float dh = __builtin_sqrtf(Bst[3]) - __builtin_sqrtf(tb.w);
  float reg = mf * (dx * dx + dy * dy + dw * dw + dh * dh);

  // ---- containing-object confidence loss ----
  float cd = best_iou - Bst[4];
  float cont = mf * cd * cd;

  // ---- block reduction ----
  float r0, r1, r2, r3;
  wave_reduce4(reg, cont, noobj, cls, r0, r1, r2, r3);
  if ((tid & 31) == 0) {
    int w = tid >> 5;
    sRed[w * 4 + 0] = r0; sRed[w * 4 + 1] = r1;
    sRed[w * 4 + 2] = r2; sRed[w * 4 + 3] = r3;
  }
  __syncthreads();
  if (tid < 4) {
    float acc = 0.0f;
    #pragma unroll
    for (int w = 0; w < 8; ++w) acc += sRed[w * 4 + tid];
    partials[(size_t)blockIdx.x * 4 + tid] = acc;
  }
}

// ---------------------------------------------------------------------------
// Finalize: reduce per-block partials, apply lambdas, write 5 outputs
// (total, reg, containing, noobj, cls) — reference return order.
// ---------------------------------------------------------------------------
__global__ void __launch_bounds__(256)
yolo_final(const float* __restrict__ partials, int nBlocks, float invN,
           float* __restrict__ out) {
  __shared__ float sRed[8 * 4];
  const int tid = threadIdx.x;
  float a0 = 0.f, a1 = 0.f, a2 = 0.f, a3 = 0.f;
  for (int r = tid; r < nBlocks; r += 256) {
    float4 p = ((const float4*)partials)[r];
    a0 += p.x; a1 += p.y; a2 += p.z; a3 += p.w;
  }
  float r0, r1, r2, r3;
  wave_reduce4(a0, a1, a2, a3, r0, r1, r2, r3);
  if ((tid & 31) == 0) {
    int w = tid >> 5;
    sRed[w * 4 + 0] = r0; sRed[w * 4 + 1] = r1;
    sRed[w * 4 + 2] = r2; sRed[w * 4 + 3] = r3;
  }
  __syncthreads();
  if (tid == 0) {
    float reg = 0.f, cont = 0.f, noobj = 0.f, cls = 0.f;
    #pragma unroll
    for (int w = 0; w < 8; ++w) {
      reg   += sRed[w * 4 + 0];
      cont  += sRed[w * 4 + 1];
      noobj += sRed[w * 4 + 2];
      cls   += sRed[w * 4 + 3];
    }
    reg   *= L_COORD;
    noobj *= L_NOOBJ;
    float total = (cls + noobj + reg + cont) * invN;
    out[0] = total;
    out[1] = reg;
    out[2] = cont;
    out[3] = noobj;
    out[4] = cls;
  }
}

extern "C" void kernel_launch(void* const* d_in, const int* in_sizes, int n_in,
                              void* d_out, int out_size, void* d_ws, size_t ws_size,
                              hipStream_t stream) {
  const float* pred          = (const float*)d_in[0];   // (N,14,14,30) f32
  const float* tbox          = (const float*)d_in[1];   // (N,14,14,4)  f32
  const float* tcls          = (const float*)d_in[2];   // (N,14,14,20) f32
  const unsigned char* hmask = (const unsigned char*)d_in[3]; // (N,14,14) bool (1B)

  const int cells   = in_sizes[3];        // N*14*14 = 802816
  const int N       = cells / 196;
  const int nBlocks = cells / CPB;        // exact: 3136

  float* partials = (float*)d_ws;         // nBlocks*4 floats = ~50 KB scratch

  yolo_main<<<nBlocks, CPB, 0, stream>>>(pred, tbox, tcls, hmask, partials);
  yolo_final<<<1, 256, 0, stream>>>(partials, nBlocks, 1.0f / (float)N,
                                    (float*)d_out);
}